// GraphConvolution_63161789055256
// MI455X (gfx1250) — compile-verified
//
#include <hip/hip_runtime.h>
#include <hip/hip_bf16.h>
#include <stdint.h>

// ---------------------------------------------------------------------------
// ST-GCN graph convolution block for MI455X (gfx1250), fp32 WMMA path.
//   y = einsum('nctv,kvw,koc->notw', x, A, W) + sum_k b   (two chained GEMMs)
//   BN over (N,T,V) per channel, then relu(y + x)
// Pass 1: fused GEMM1(W_k@x) -> GEMM2(z@A_k) with v_wmma_f32_16x16x4_f32.
//         x tile staged via global_load_async_to_lds_b64 (ASYNCcnt path).
//         Writes pre-BN y + deterministic per-block channel sums.
// Pass 2: deterministic tree reduction of stats.
// Pass 3: float4 BN + residual + relu.
// ---------------------------------------------------------------------------

typedef float v2f __attribute__((ext_vector_type(2)));
typedef float v8f __attribute__((ext_vector_type(8)));

#define NN 64
#define CC 64
#define TT 300
#define VV 25
#define KK 3
#define TB 6                    // timesteps per block
#define LCOLS (TB * VV)         // 150 columns of (t,v) per block
#define NTB (TT / TB)           // 50 t-blocks
#define NBLK (NN * NTB)         // 3200 workgroups in pass 1
#define XS 168                  // xs row stride (>=160, bank-safe)
#define ZS (TB * 32 + 4)        // 196: v padded to 32 per t, +4 anti-conflict
#define APS 33                  // A_pad row stride (32 + 1)
#define WS 68                   // W_k row stride (64 + 4, conflict-free)
#define EPSF 1e-5f
#define CNTF ((float)(NN * TT * VV))   // 480000 per-channel count

// LDS layout (float offsets)
#define OFF_XS   0
#define OFF_ZS   (OFF_XS + CC * XS)            // 10752
#define OFF_AP   (OFF_ZS + CC * ZS)            // 23296
#define OFF_WK   (OFF_AP + KK * 32 * APS)      // 26464
#define OFF_SB   (OFF_WK + CC * WS)            // 30816
#define OFF_RS   (OFF_SB + CC)                 // 30880  lred_sum[64][12]
#define OFF_RQ   (OFF_RS + CC * 12)            // 31648  lred_sq [64][12]
#define LDS_FLOATS (OFF_RQ + CC * 12)          // 32416
#define LDS_BYTES (LDS_FLOATS * 4)             // 129664 B (<320KB WGP budget)

// workspace layout (float offsets into d_ws)
#define WSO_GSUM 0
#define WSO_GSQ  64
#define WSO_PSUM 128
#define WSO_PSQ  (WSO_PSUM + NBLK * CC)        // 128 + 204800
#define WSO_Y    (WSO_PSQ + NBLK * CC)         // 409728

extern "C" __global__ __launch_bounds__(192) void stgcn_fused(
    const float* __restrict__ x, const float* __restrict__ A,
    const float* __restrict__ W, const float* __restrict__ b,
    float* __restrict__ yws, float* __restrict__ psum, float* __restrict__ psq)
{
  extern __shared__ float lds[];
  float* xs   = lds + OFF_XS;
  float* zs   = lds + OFF_ZS;
  float* ap   = lds + OFF_AP;
  float* wk   = lds + OFF_WK;
  float* sb   = lds + OFF_SB;
  float* lrs  = lds + OFF_RS;
  float* lrq  = lds + OFF_RQ;

  const int tid  = threadIdx.x;
  const int lane = tid & 31;
  const int wv   = tid >> 5;       // wave 0..5, owns t-local = wv in stage 2
  const int hlf  = lane >> 4;      // half-wave 0/1
  const int l16  = lane & 15;

  const int bid = blockIdx.x;
  const int n   = bid / NTB;
  const int tb  = bid % NTB;

  // ---- stage x tile via async global->LDS (B64, 8B-aligned), zero pads ----
  {
    const float* xb = x + (size_t)n * CC * TT * VV + (size_t)tb * LCOLS;
    // 64 rows x 75 float2 pairs: memory -> LDS without VGPR round trip
    for (int p = tid; p < CC * (LCOLS / 2); p += 192) {
      int c = p / (LCOLS / 2), col2 = p - c * (LCOLS / 2);
      uint32_t ldsb = (uint32_t)(uintptr_t)(xs + c * XS + col2 * 2);
      uint32_t voff = (uint32_t)((c * TT * VV + col2 * 2) * 4);
      asm volatile("global_load_async_to_lds_b64 %0, %1, %2"
                   :: "v"(ldsb), "v"(voff), "s"(xb) : "memory");
    }
    // zero xs padding columns (read by ragged stage-1 col tiles)
    for (int idx = tid; idx < CC * (XS - LCOLS); idx += 192) {
      int c = idx / (XS - LCOLS), col = LCOLS + idx % (XS - LCOLS);
      xs[c * XS + col] = 0.0f;
    }
    // zero zs entirely: stage 2 reads padded v=25..31 which stage 1 never
    // writes; garbage NaNs there would poison acc (NaN * 0 = NaN).
    for (int idx = tid; idx < CC * ZS; idx += 192) zs[idx] = 0.0f;
    for (int idx = tid; idx < KK * 32 * APS; idx += 192) ap[idx] = 0.0f;
    asm volatile("s_wait_asynccnt 0" ::: "memory");
  }
  __syncthreads();
  // ---- fill A_pad (25x25 -> 32x33 zero-padded) and bias sum ----
  for (int idx = tid; idx < KK * VV * VV; idx += 192) {
    int k = idx / (VV * VV);
    int r = idx - k * VV * VV;
    int v = r / VV, w = r - v * VV;
    ap[(k * 32 + v) * APS + w] = A[idx];
  }
  if (tid < CC) sb[tid] = b[tid] + b[CC + tid] + b[2 * CC + tid];

  v8f acc2[4][2];
#pragma unroll
  for (int r = 0; r < 4; ++r)
#pragma unroll
    for (int h = 0; h < 2; ++h) acc2[r][h] = (v8f){0,0,0,0,0,0,0,0};

  for (int k = 0; k < KK; ++k) {
    __syncthreads();
    // stage W[k] (koc layout): wk[o*WS + c]
    for (int idx = tid; idx < CC * CC; idx += 192) {
      int o = idx >> 6, c = idx & 63;
      wk[o * WS + c] = W[(size_t)k * CC * CC + idx];
    }
    __syncthreads();

    // ---- stage 1: zs = W_k (64x64) @ xs (64x150), f32 WMMA K=4 ----
    for (int t = wv; t < 40; t += 6) {         // 4 o-tiles x 10 col-tiles
      int r = t & 3, ct = t >> 2;
      int ncol = ct * 16 + l16;
      v8f acc = {0,0,0,0,0,0,0,0};
#pragma unroll
      for (int j = 0; j < 16; ++j) {
        int krow = 4 * j + 2 * hlf;
        // A operand: W[o][c], M=l16 (o-row), K-chunk per ISA layout
        v2f a = *(const v2f*)&wk[(16 * r + l16) * WS + krow];
        // B operand: xs[c][col], K rows krow/krow+1
        v2f bb;
        bb.x = xs[(krow    ) * XS + ncol];
        bb.y = xs[(krow + 1) * XS + ncol];
        acc = __builtin_amdgcn_wmma_f32_16x16x4_f32(
            false, a, false, bb, (short)0, acc, false, false);
      }
      if (ncol < LCOLS) {
        int tl = ncol / VV, vv = ncol - tl * VV;
        int zc = tl * 32 + vv;                 // v padded to 32 per t
#pragma unroll
        for (int q = 0; q < 8; ++q) {
          int o = 16 * r + q + 8 * hlf;        // C/D layout row
          zs[o * ZS + zc] = acc[q];
        }
      }
    }
    __syncthreads();

    // ---- stage 2: acc2 += zs (64 x 32-per-t) @ A_pad[k] (32x32) ----
    {
      const int tl = wv;                       // one t-local per wave
#pragma unroll
      for (int r = 0; r < 4; ++r) {
#pragma unroll
        for (int h = 0; h < 2; ++h) {
          v8f c = acc2[r][h];
#pragma unroll
          for (int j = 0; j < 8; ++j) {
            int v0 = 4 * j + 2 * hlf;
            v2f a = *(const v2f*)&zs[(16 * r + l16) * ZS + tl * 32 + v0];
            v2f bb;
            bb.x = ap[(k * 32 + v0    ) * APS + 16 * h + l16];
            bb.y = ap[(k * 32 + v0 + 1) * APS + 16 * h + l16];
            c = __builtin_amdgcn_wmma_f32_16x16x4_f32(
                false, a, false, bb, (short)0, c, false, false);
          }
          acc2[r][h] = c;
        }
      }
    }
  }

  // ---- epilogue: + bias, write pre-BN y, deterministic channel stats ----
  {
    const int tl = wv;
    const int t = tb * TB + tl;
#pragma unroll
    for (int r = 0; r < 4; ++r) {
#pragma unroll
      for (int h = 0; h < 2; ++h) {
        v8f c = acc2[r][h];
        int w = 16 * h + l16;
        bool okw = (w < VV);
#pragma unroll
        for (int q = 0; q < 8; ++q) {
          int o = 16 * r + q + 8 * hlf;
          float val = c[q] + sb[o];
          if (okw) yws[(((size_t)n * CC + o) * TT + t) * VV + w] = val;
          float s  = okw ? val : 0.0f;
          float s2 = s * s;
#pragma unroll
          for (int m = 1; m <= 8; m <<= 1) {   // fixed tree over 16 lanes
            s  += __shfl_xor(s,  m, 32);
            s2 += __shfl_xor(s2, m, 32);
          }
          if (l16 == 0) {                      // unique (o, tl, h) slot
            lrs[o * 12 + tl * 2 + h] = s;
            lrq[o * 12 + tl * 2 + h] = s2;
          }
        }
      }
    }
  }
  __syncthreads();
  if (tid < CC) {
    float s = 0.0f, s2 = 0.0f;
#pragma unroll
    for (int i = 0; i < 12; ++i) { s += lrs[tid * 12 + i]; s2 += lrq[tid * 12 + i]; }
    psum[(size_t)bid * CC + tid] = s;
    psq [(size_t)bid * CC + tid] = s2;
  }
}

extern "C" __global__ __launch_bounds__(256) void stgcn_reduce(
    const float* __restrict__ psum, const float* __restrict__ psq,
    float* __restrict__ gsum, float* __restrict__ gsq)
{
  __shared__ float sh0[256], sh1[256];
  const int c = blockIdx.x;            // one channel per block
  const int tid = threadIdx.x;
  float s = 0.0f, s2 = 0.0f;
  for (int i = tid; i < NBLK; i += 256) {
    s  += psum[(size_t)i * CC + c];
    s2 += psq [(size_t)i * CC + c];
  }
  sh0[tid] = s; sh1[tid] = s2;
  __syncthreads();
  for (int st = 128; st > 0; st >>= 1) {
    if (tid < st) { sh0[tid] += sh0[tid + st]; sh1[tid] += sh1[tid + st]; }
    __syncthreads();
  }
  if (tid == 0) { gsum[c] = sh0[0]; gsq[c] = sh1[0]; }
}

extern "C" __global__ __launch_bounds__(256) void stgcn_finalize(
    const float* __restrict__ yws, const float* __restrict__ x,
    const float* __restrict__ gsum, const float* __restrict__ gsq,
    const float* __restrict__ gamma, const float* __restrict__ beta,
    float* __restrict__ out)
{
  const float inv = 1.0f / CNTF;
  const int total4 = NN * CC * TT * VV / 4;       // 7,680,000
  const int rowlen4 = TT * VV / 4;                // 1875
  for (int i = blockIdx.x * blockDim.x + threadIdx.x; i < total4;
       i += gridDim.x * blockDim.x) {
    int row = i / rowlen4;
    int c = row & 63;
    float mean  = gsum[c] * inv;
    float var   = gsq[c] * inv - mean * mean;
    float scale = gamma[c] * rsqrtf(var + EPSF);
    float shift = beta[c] - mean * scale;
    float4 y  = ((const float4*)yws)[i];
    float4 xv = ((const float4*)x)[i];
    float4 o;
    o.x = fmaxf(fmaf(y.x, scale, shift) + xv.x, 0.0f);
    o.y = fmaxf(fmaf(y.y, scale, shift) + xv.y, 0.0f);
    o.z = fmaxf(fmaf(y.z, scale, shift) + xv.z, 0.0f);
    o.w = fmaxf(fmaf(y.w, scale, shift) + xv.w, 0.0f);
    ((float4*)out)[i] = o;
  }
}

extern "C" void kernel_launch(void* const* d_in, const int* in_sizes, int n_in,
                              void* d_out, int out_size, void* d_ws, size_t ws_size,
                              hipStream_t stream) {
  const float* x     = (const float*)d_in[0];
  const float* A     = (const float*)d_in[1];
  const float* W     = (const float*)d_in[2];
  const float* b     = (const float*)d_in[3];
  const float* gamma = (const float*)d_in[4];
  const float* beta  = (const float*)d_in[5];

  float* ws    = (float*)d_ws;
  float* gsum  = ws + WSO_GSUM;
  float* gsq   = ws + WSO_GSQ;
  float* psum  = ws + WSO_PSUM;
  float* psq   = ws + WSO_PSQ;
  float* yws   = ws + WSO_Y;

  (void)hipFuncSetAttribute(reinterpret_cast<const void*>(stgcn_fused),
                            hipFuncAttributeMaxDynamicSharedMemorySize,
                            LDS_BYTES);

  stgcn_fused<<<dim3(NBLK), dim3(192), LDS_BYTES, stream>>>(
      x, A, W, b, yws, psum, psq);
  stgcn_reduce<<<dim3(CC), dim3(256), 0, stream>>>(psum, psq, gsum, gsq);
  stgcn_finalize<<<dim3(2048), dim3(256), 0, stream>>>(
      yws, x, gsum, gsq, gamma, beta, (float*)d_out);
}